// t_Conv_47699906789847
// MI455X (gfx1250) — compile-verified
//
#include <hip/hip_runtime.h>
#include <math.h>

typedef float v2f __attribute__((ext_vector_type(2)));
typedef float v8f __attribute__((ext_vector_type(8)));

#define MAX_SPIKE 100.0f
#define FIN 144        // 16 channels * 3*3
#define FOUT 64
#define CH 16
#define HH 32
#define WW 32
#define WQ_STRIDE 68   // padded LDS row stride (floats) to break bank conflicts
#define WAVES 8        // 8 wave32s per block, one spatial position per wave

// ---------------------------------------------------------------------------
// Kernel 1: weight quantization (sort-independent, hoisted out of the hot loop)
//   wq[o][i] = round(clip(tanh(w)/alpha,-1,1)*127) * alpha/127,
//   alpha = max|tanh(w)|  (global scalar).  Stored transposed: wqT[i*64+o].
// ---------------------------------------------------------------------------
__global__ __launch_bounds__(256) void quant_weights(const float* __restrict__ w,
                                                     float* __restrict__ wqT) {
  __shared__ float red[256];
  const int tid = threadIdx.x;
  float m = 0.0f;
  for (int i = tid; i < FOUT * FIN; i += 256) m = fmaxf(m, fabsf(tanhf(w[i])));
  red[tid] = m;
  __syncthreads();
  for (int s = 128; s > 0; s >>= 1) {
    if (tid < s) red[tid] = fmaxf(red[tid], red[tid + s]);
    __syncthreads();
  }
  const float alpha = red[0];
  for (int i = tid; i < FOUT * FIN; i += 256) {
    int o = i / FIN, f = i - o * FIN;
    float t = tanhf(w[i]);
    float q = rintf(fminf(fmaxf(t / alpha, -1.0f), 1.0f) * 127.0f);  // RNE, like jnp.round
    wqT[f * FOUT + o] = q * alpha * (1.0f / 127.0f);
  }
}

// ---------------------------------------------------------------------------
// Kernel 2: 8 wave32s per block, one spatial position per wave; the quantized
// weight matrix is staged into LDS ONCE per block (float4 -> ds_store_b128)
// and shared by all 8 waves.  Per wave: stable rank sort of the 144 spike
// times, then prefix sums along the sorted axis as blocked lower-triangular
// matmuls on V_WMMA_F32_16X16X4_F32, then the spike-time min reduction.
// ---------------------------------------------------------------------------
__global__ __launch_bounds__(256) void snn_conv(const float* __restrict__ inp,
                                                const float* __restrict__ wqT,
                                                float* __restrict__ out) {
  __shared__ float s_wq[FIN * WQ_STRIDE];       // one shared copy per block
  __shared__ float s_vals[WAVES][FIN];          // raw spike times, per wave
  __shared__ float s_si[WAVES][FIN + 1];        // sorted times (+1.0 sentinel)
  __shared__ int   s_idx[WAVES][FIN];           // argsort, per wave

  const int tid    = threadIdx.x;
  const int waveId = tid >> 5;
  const int lane   = tid & 31;
  const int laneLo = lane & 15;
  const int laneHi = lane >> 4;

  const int pos = blockIdx.x * WAVES + waveId;
  const int b = pos >> 10;                      // H*W = 1024
  const int l = pos & 1023;
  const int h = l >> 5, w = l & 31;

  // ---- cooperative weight staging: 2304 float4s over 256 threads ----------
  {
    const float4* wq4 = (const float4*)wqT;
    for (int i = tid; i < (FIN * FOUT) / 4; i += 256) {
      int e = i * 4;
      int f = e >> 6, o = e & 63;               // 64 % 4 == 0 -> stays in-row
      *(float4*)&s_wq[f * WQ_STRIDE + o] = wq4[i];
    }
  }

  // ---- per-wave patch gather: pad -> 0 -> masked; < 0.1 -> MAX_SPIKE ------
  for (int f = lane; f < FIN; f += 32) {
    int c = f / 9, rr = (f % 9) / 3, ss = f % 3;
    int ph = h + rr - 1, pw = w + ss - 1;
    float v = 0.0f;
    if (ph >= 0 && ph < HH && pw >= 0 && pw < WW)
      v = inp[((b * CH + c) * HH + ph) * WW + pw];
    if (v < 0.1f) v = MAX_SPIKE;
    s_vals[waveId][f] = v;
  }
  __syncthreads();

  // ---- stable rank sort: rank[f] = #{j : v[j]<v[f] || (v[j]==v[f] && j<f)} -
  {
    float myv[5]; int myf[5]; int cnt[5];
#pragma unroll
    for (int t = 0; t < 5; ++t) {
      myf[t] = lane + 32 * t;
      myv[t] = (myf[t] < FIN) ? s_vals[waveId][myf[t]] : 0.0f;
      cnt[t] = 0;
    }
    for (int j = 0; j < FIN; ++j) {
      float vj = s_vals[waveId][j];             // LDS broadcast within the wave
#pragma unroll
      for (int t = 0; t < 5; ++t)
        cnt[t] += (vj < myv[t]) || (vj == myv[t] && j < myf[t]);
    }
#pragma unroll
    for (int t = 0; t < 5; ++t)
      if (myf[t] < FIN) { s_si[waveId][cnt[t]] = myv[t]; s_idx[waveId][cnt[t]] = myf[t]; }
    if (lane == 0) s_si[waveId][FIN] = 1.0f;    // causality sentinel for last slot
  }
  __syncthreads();

  // ---- constant triangular A blocks: A[m][j] = (j <= m), 4 K=4 steps ------
  // A layout (16x4 f32): lane -> M = lane&15 ; VGPR v -> K = v + 2*(lane>>4)
  v2f a4[4];
#pragma unroll
  for (int ks = 0; ks < 4; ++ks)
#pragma unroll
    for (int v = 0; v < 2; ++v) {
      int col = 4 * ks + v + 2 * laneHi;
      a4[ks][v] = (col <= laneLo) ? 1.0f : 0.0f;
    }

  float carryS[4] = {0, 0, 0, 0};   // running column totals of ws      (per N-tile)
  float carryT[4] = {0, 0, 0, 0};   // running column totals of si*ws   (per N-tile)
  float mn[4] = {MAX_SPIKE, MAX_SPIKE, MAX_SPIKE, MAX_SPIKE};

  for (int kt = 0; kt < 9; ++kt) {
    const int kbase = kt * 16;
#pragma unroll
    for (int nt = 0; nt < 4; ++nt) {
      v8f accS = {0, 0, 0, 0, 0, 0, 0, 0};
      v8f accT = {0, 0, 0, 0, 0, 0, 0, 0};
#pragma unroll
      for (int ks = 0; ks < 4; ++ks) {
        // B layout (4x16 f32): lane -> N = lane&15 ; VGPR v -> K = v + 2*(lane>>4)
        int k0 = kbase + 4 * ks + 2 * laneHi;
        int i0 = s_idx[waveId][k0], i1 = s_idx[waveId][k0 + 1];
        float si0 = s_si[waveId][k0], si1 = s_si[waveId][k0 + 1];
        int col = nt * 16 + laneLo;
        v2f bw, bs;
        bw.x = s_wq[i0 * WQ_STRIDE + col];
        bw.y = s_wq[i1 * WQ_STRIDE + col];
        bs.x = bw.x * si0;
        bs.y = bw.y * si1;
        accS = __builtin_amdgcn_wmma_f32_16x16x4_f32(false, a4[ks], false, bw,
                                                     (short)0, accS, false, false);
        accT = __builtin_amdgcn_wmma_f32_16x16x4_f32(false, a4[ks], false, bs,
                                                     (short)0, accT, false, false);
      }
      // ---- spike-time evaluation for the 16 sorted slots of this tile -----
#pragma unroll
      for (int v = 0; v < 8; ++v) {
        int m = v + 8 * laneHi;                 // C layout: VGPR v -> row v / v+8
        int k = kbase + m;
        float wsum = accS[v] + carryS[nt];
        float iws  = accT[v] + carryT[nt];
        float sik  = s_si[waveId][k];
        float nxt  = s_si[waveId][k + 1];
        float den  = fminf(fmaxf(wsum - 1.0f, 1e-10f), 1e10f);
        float sp   = iws / den;
        if (wsum < 1.0f) sp = MAX_SPIKE;
        if (sp < sik)    sp = MAX_SPIKE;
        if (sp > nxt)    sp = MAX_SPIKE;
        mn[nt] = fminf(mn[nt], sp);
      }
      // carry += tile column totals = in-tile cumsum row 15 (VGPR 7, lanes 16..31)
      carryS[nt] += __shfl(accS[7], 16 + laneLo, 32);
      carryT[nt] += __shfl(accT[7], 16 + laneLo, 32);
    }
  }

  // ---- combine the two half-wave row groups, then store -------------------
#pragma unroll
  for (int nt = 0; nt < 4; ++nt) {
    float other = __shfl(mn[nt], lane ^ 16, 32);
    mn[nt] = fminf(mn[nt], other);
  }
  if (lane < 16) {
#pragma unroll
    for (int nt = 0; nt < 4; ++nt) {
      int o = nt * 16 + laneLo;
      out[((b * FOUT + o) * HH + h) * WW + w] = mn[nt];
    }
  }
}

// ---------------------------------------------------------------------------
extern "C" void kernel_launch(void* const* d_in, const int* in_sizes, int n_in,
                              void* d_out, int out_size, void* d_ws, size_t ws_size,
                              hipStream_t stream) {
  (void)n_in; (void)out_size; (void)ws_size;
  const float* inp = (const float*)d_in[0];
  const float* wt  = (const float*)d_in[1];
  float* out = (float*)d_out;
  float* wqT = (float*)d_ws;                    // 144*64 floats of scratch

  const int B = in_sizes[0] / (CH * HH * WW);
  const int positions = B * HH * WW;            // 1024 per batch -> divisible by 8

  quant_weights<<<1, 256, 0, stream>>>(wt, wqT);
  snn_conv<<<positions / WAVES, 256, 0, stream>>>(inp, wqT, out);
}